// ClassCapsules_46926812676682
// MI455X (gfx1250) — compile-verified
//
#include <hip/hip_runtime.h>

typedef __attribute__((ext_vector_type(2))) float v2f;
typedef __attribute__((ext_vector_type(8))) float v8f;

#define R_CAPS 1152
#define C_CAPS 10
#define O_DIM  16
#define I_DIM  8
#define B_SZ   256
#define K_RI   (R_CAPS * I_DIM)   // 9216
#define N_CO   (C_CAPS * O_DIM)   // 160

// -------------------------------------------------------------------------
// Kernel 1: c_ij = softmax(b_ij, axis=C); Weff[(r,i),(c,o)] = c_ij[r,c]*W[r,c,o,i]
// One block per r.
// -------------------------------------------------------------------------
__global__ void __launch_bounds__(128)
softmax_weff_kernel(const float* __restrict__ W,
                    const float* __restrict__ b_ij,
                    float* __restrict__ Weff) {
  int r = blockIdx.x;
  float logits[C_CAPS];
  float mx = -1e30f;
  #pragma unroll
  for (int c = 0; c < C_CAPS; ++c) {
    logits[c] = b_ij[r * C_CAPS + c];
    mx = fmaxf(mx, logits[c]);
  }
  float sum = 0.f;
  #pragma unroll
  for (int c = 0; c < C_CAPS; ++c) {
    logits[c] = __expf(logits[c] - mx);
    sum += logits[c];
  }
  float inv = 1.f / sum;
  const float* Wr = W + (size_t)r * (C_CAPS * O_DIM * I_DIM);
  for (int e = threadIdx.x; e < C_CAPS * O_DIM * I_DIM; e += 128) {
    int c = e >> 7;          // / (O*I)
    int o = (e >> 3) & 15;
    int i = e & 7;
    float val = logits[c] * inv * Wr[e];
    Weff[(size_t)(r * I_DIM + i) * N_CO + c * O_DIM + o] = val;
  }
}

// -------------------------------------------------------------------------
// Kernel 2: S = X[256,9216] @ Weff[9216,160]; fused bias + squash -> V[256,160]
// grid.x = 16 (M tiles of 16 batch rows); threadIdx.y = capsule c (N tile).
// One wave per 16x16 D tile, K = 9216 via v_wmma_f32_16x16x4_f32.
// -------------------------------------------------------------------------
__global__ void __launch_bounds__(320)
s_v_kernel(const float* __restrict__ X,
           const float* __restrict__ Weff,
           const float* __restrict__ bias,
           float* __restrict__ Vout) {
  int lane = threadIdx.x;        // 0..31
  int c    = threadIdx.y;        // 0..9
  int m0   = blockIdx.x * 16;
  int half = lane >> 4;
  int l16  = lane & 15;

  // A frag: lane holds A[m0+l16, k0 + 2*half + {0,1}]
  const float* Xrow = X + (size_t)(m0 + l16) * K_RI + 2 * half;
  // B frag: lane holds B[k0 + 2*half + {0,1}, c*16 + l16]
  const float* Bcol = Weff + (size_t)(2 * half) * N_CO + c * O_DIM + l16;

  v8f acc = {};
  #pragma unroll 8
  for (int k0 = 0; k0 < K_RI; k0 += 4) {
    v2f a = { Xrow[k0], Xrow[k0 + 1] };
    v2f b = { Bcol[(size_t)k0 * N_CO], Bcol[(size_t)(k0 + 1) * N_CO] };
    acc = __builtin_amdgcn_wmma_f32_16x16x4_f32(false, a, false, b,
                                                (short)0, acc, false, false);
  }

  // bias depends only on column (o = l16)
  float bv = bias[c * O_DIM + l16];
  float vals[8];
  #pragma unroll
  for (int v = 0; v < 8; ++v) vals[v] = acc[v] + bv;

  // squash: n2 = row-sum over the 16 columns (lanes within each half)
  #pragma unroll
  for (int v = 0; v < 8; ++v) {
    float s2 = vals[v] * vals[v];
    s2 += __shfl_xor(s2, 1);
    s2 += __shfl_xor(s2, 2);
    s2 += __shfl_xor(s2, 4);
    s2 += __shfl_xor(s2, 8);
    float scale = s2 / ((1.f + s2) * sqrtf(s2 + 1e-8f));
    int m = m0 + half * 8 + v;   // D row: lanes 0-15 -> m0+v, lanes 16-31 -> m0+8+v
    Vout[(size_t)m * N_CO + c * O_DIM + l16] = vals[v] * scale;
  }
}

// -------------------------------------------------------------------------
// Kernel 3: T^T[co, ri] = sum_b V[b,co] * X[b,ri]  (M=160, N=9216, K=256),
// fused with agree[r,c] = (1/B) * sum_{o,i} W[r,c,o,i]*T and b_ij += agree.
// grid = (10 capsules = M tiles, 576 ri tiles). One wave per 16x16 tile.
// -------------------------------------------------------------------------
__global__ void __launch_bounds__(32)
agree_kernel(const float* __restrict__ X,
             const float* __restrict__ V,
             const float* __restrict__ W,
             float* __restrict__ b_ij) {
  int lane = threadIdx.x;
  int c    = blockIdx.x;          // M tile: rows co = c*16 .. c*16+15
  int nt   = blockIdx.y;          // N tile: cols ri = nt*16 .. nt*16+15
  int half = lane >> 4;
  int l16  = lane & 15;
  int m0 = c * 16;
  int n0 = nt * 16;

  // A[m=co, k=b] = V[b*160 + co]  (strided);  B[k=b, n=ri] = X[b*9216 + ri]
  const float* Acol = V + (size_t)(2 * half) * N_CO + m0 + l16;
  const float* Bcol = X + (size_t)(2 * half) * K_RI + n0 + l16;

  v8f acc = {};
  #pragma unroll 8
  for (int k0 = 0; k0 < B_SZ; k0 += 4) {
    v2f a = { Acol[(size_t)k0 * N_CO], Acol[(size_t)(k0 + 1) * N_CO] };
    v2f b = { Bcol[(size_t)k0 * K_RI], Bcol[(size_t)(k0 + 1) * K_RI] };
    acc = __builtin_amdgcn_wmma_f32_16x16x4_f32(false, a, false, b,
                                                (short)0, acc, false, false);
  }

  // lane covers column ri = n0 + l16 -> (r, i); rows o = half*8 + v
  int r = nt * 2 + (l16 >> 3);
  int i = l16 & 7;
  const float* Wp = W + (size_t)((r * C_CAPS + c) * O_DIM + half * 8) * I_DIM + i;
  float partial = 0.f;
  #pragma unroll
  for (int v = 0; v < 8; ++v) partial += acc[v] * Wp[(size_t)v * I_DIM];

  // reduce: i within each 8-lane group, then combine the two o-halves
  partial += __shfl_xor(partial, 1);
  partial += __shfl_xor(partial, 2);
  partial += __shfl_xor(partial, 4);
  partial += __shfl_xor(partial, 16);

  // lane 0 holds full sum for r0 = 2*nt, lane 8 for r1 = 2*nt+1 (unique writer)
  if (lane == 0 || lane == 8) {
    int idx = r * C_CAPS + c;
    b_ij[idx] = b_ij[idx] + partial * (1.0f / 256.0f);
  }
}

// -------------------------------------------------------------------------
extern "C" void kernel_launch(void* const* d_in, const int* in_sizes, int n_in,
                              void* d_out, int out_size, void* d_ws, size_t ws_size,
                              hipStream_t stream) {
  (void)in_sizes; (void)n_in; (void)out_size; (void)ws_size;
  const float* x    = (const float*)d_in[0];   // [256,1152,8]
  const float* W    = (const float*)d_in[1];   // [1152,10,16,8]
  const float* bias = (const float*)d_in[2];   // [10,16]
  float* out = (float*)d_out;                  // [256,10,16,1]

  char* ws = (char*)d_ws;
  float* Weff = (float*)ws;                                  // 9216*160*4 = 5.9 MB
  float* Vbuf = (float*)(ws + 6u * 1024u * 1024u);           // 256*160*4  = 160 KB
  float* b_ij = (float*)(ws + 6u * 1024u * 1024u + 256u * 1024u); // 1152*10*4 = 45 KB

  hipMemsetAsync(b_ij, 0, R_CAPS * C_CAPS * sizeof(float), stream);

  for (int it = 0; it < 3; ++it) {
    softmax_weff_kernel<<<R_CAPS, 128, 0, stream>>>(W, b_ij, Weff);
    float* vdst = (it == 2) ? out : Vbuf;
    s_v_kernel<<<dim3(16, 1, 1), dim3(32, 10, 1), 0, stream>>>(x, Weff, bias, vdst);
    if (it < 2)
      agree_kernel<<<dim3(10, 576, 1), dim3(32, 1, 1), 0, stream>>>(x, Vbuf, W, b_ij);
  }
}